// MorphologicalLayer_6871947673664
// MI455X (gfx1250) — compile-verified
//
#include <hip/hip_runtime.h>
#include <hip/hip_bf16.h>

// Morphological chain: closing(k=51) then opening(k=25) over rows of length L,
// zero-padded 'same' per stage. Fused single kernel, log-doubling sliding
// min/max in LDS, async B128 global->LDS fill (CDNA5 GLOBAL_LOAD_ASYNC_TO_LDS_B128).

#define T_TILE   2048
#define HALO     80                   // >= 74 receptive-field growth; 16B-aligned halo
#define NBUF     (T_TILE + 2 * HALO)  // 2208 floats per buffer (16B multiple)
#define NTHREADS 256

// ---- CDNA5 async global->LDS path (guarded; falls back to plain LDS stores) ----
#if defined(__has_builtin)
#if defined(__gfx1250__) && __has_builtin(__builtin_amdgcn_s_wait_asynccnt)
#if __has_builtin(__builtin_amdgcn_global_load_async_to_lds_b128)
#define USE_ASYNC_B128 1
#elif __has_builtin(__builtin_amdgcn_global_load_async_to_lds_b32)
#define USE_ASYNC_B32 1
#endif
#endif
#endif

#if defined(USE_ASYNC_B128) || defined(USE_ASYNC_B32)
typedef int v4i __attribute__((vector_size(16)));         // 4 x i32
typedef __attribute__((address_space(1))) int gas_int;    // global (as1) scalar
typedef __attribute__((address_space(3))) int las_int;    // LDS    (as3) scalar
typedef __attribute__((address_space(1))) v4i gas_v4i;    // global (as1) vector
typedef __attribute__((address_space(3))) v4i las_v4i;    // LDS    (as3) vector
#define HAVE_ASYNC 1
#endif

// Copy 4 consecutive floats global -> LDS (both 16B aligned).
__device__ __forceinline__ void fill_vec4(const float* gsrc, float* lds_dst) {
#if defined(USE_ASYNC_B128)
    gas_v4i* gp = (gas_v4i*)(unsigned long long)gsrc;
    las_v4i* lp = (las_v4i*)(unsigned long long)lds_dst;
    __builtin_amdgcn_global_load_async_to_lds_b128(gp, lp, /*offset=*/0, /*cpol=*/0);
#elif defined(USE_ASYNC_B32)
    gas_int* gp = (gas_int*)(unsigned long long)gsrc;
    las_int* lp = (las_int*)(unsigned long long)lds_dst;
    __builtin_amdgcn_global_load_async_to_lds_b32(gp + 0, lp + 0, 0, 0);
    __builtin_amdgcn_global_load_async_to_lds_b32(gp + 1, lp + 1, 0, 0);
    __builtin_amdgcn_global_load_async_to_lds_b32(gp + 2, lp + 2, 0, 0);
    __builtin_amdgcn_global_load_async_to_lds_b32(gp + 3, lp + 3, 0, 0);
#else
    *(float4*)lds_dst = *(const float4*)gsrc;
#endif
}

__device__ __forceinline__ void wait_async_fill() {
#ifdef HAVE_ASYNC
    __builtin_amdgcn_s_wait_asynccnt(0);
#endif
}

// One sliding min/max stage of odd window k (k in {51,25}) via log-doubling.
// cur holds input valid on [lo, hi). On return cur holds output valid on the
// updated [lo, hi), with exact 0.0 forced at global-OOB positions (zero pad).
template <bool IS_MAX>
__device__ __forceinline__ void sliding_stage(float*& cur, float*& oth,
                                              int& lo, int& hi,
                                              const int k, const int base,
                                              const int t0, const int L,
                                              const int tid)
{
    float* s = cur;
    float* d = oth;
    int vhi = hi;
    // Build m_base: window-of-2^p maxima/minima, ping-ponging s<->d.
    for (int w = 1; w < base; w <<= 1) {
        for (int b = lo + tid; b + w < vhi; b += NTHREADS) {
            const float a0 = s[b];
            const float a1 = s[b + w];
            d[b] = IS_MAX ? fmaxf(a0, a1) : fminf(a0, a1);
        }
        __syncthreads();
        vhi -= w;
        float* t = s; s = d; d = t;
    }
    // Combine two overlapping base-windows into window-k, apply zero padding.
    const int off  = k - base;          // 51-32=19, 25-16=9
    const int left = (k - 1) >> 1;      // 25, 12
    const int nlo  = lo + left;
    const int nhi  = vhi - off + left;  // == hi - k + 1 + left
    for (int p = nlo + tid; p < nhi; p += NTHREADS) {
        const int j = p - left;
        const float a0 = s[j];
        const float a1 = s[j + off];
        float v = IS_MAX ? fmaxf(a0, a1) : fminf(a0, a1);
        const int g = t0 - HALO + p;    // global position of buffer slot p
        d[p] = (g < 0 || g >= L) ? 0.0f : v;
    }
    __syncthreads();
    cur = d; oth = s;
    lo = nlo; hi = nhi;
}

__global__ __launch_bounds__(NTHREADS)
void morph_co_51_25_kernel(const float* __restrict__ x,
                           float* __restrict__ out,
                           const int L)
{
    __shared__ __align__(16) float bufP[NBUF];
    __shared__ __align__(16) float bufQ[NBUF];

    const int tid = threadIdx.x;
    const int t0  = blockIdx.x * T_TILE;           // first output index of tile
    const long long rowBase = (long long)blockIdx.y * (long long)L;
    const float* __restrict__ xrow = x + rowBase;

    // Speculative prefetch of the next tile's head (emits global_prefetch_b8).
    if (tid == 0 && t0 + T_TILE < L) {
        __builtin_prefetch((const void*)(xrow + t0 + T_TILE), 0, 0);
    }

    // Fill bufP with x[t0-HALO .. t0+T_TILE+HALO), zeros outside [0, L).
    // t0 is a multiple of T_TILE and HALO is a multiple of 4, so every 4-float
    // vector of the load region is 16B-aligned and fully in- or out-of-range.
    for (int v = tid; v < NBUF / 4; v += NTHREADS) {
        const int b = 4 * v;
        const int g = t0 - HALO + b;
        if (g >= 0 && g < L) {
            fill_vec4(xrow + g, &bufP[b]);
        } else {
            *(float4*)&bufP[b] = make_float4(0.0f, 0.0f, 0.0f, 0.0f);
        }
    }
    wait_async_fill();
    __syncthreads();

    float* cur = bufP;
    float* oth = bufQ;
    int lo = 0, hi = NBUF;

    // closing, k=51: dilation (max) then erosion (min)
    sliding_stage<true >(cur, oth, lo, hi, 51, 32, t0, L, tid);
    sliding_stage<false>(cur, oth, lo, hi, 51, 32, t0, L, tid);
    // opening, k=25: erosion (min) then dilation (max)
    sliding_stage<false>(cur, oth, lo, hi, 25, 16, t0, L, tid);
    sliding_stage<true >(cur, oth, lo, hi, 25, 16, t0, L, tid);
    // cur now valid on [74, T+86) which covers outputs at [HALO, HALO+T_TILE).

    // Vectorized store of outputs [t0, t0+T_TILE) (both sides 16B-aligned).
    for (int v = tid; v < T_TILE / 4; v += NTHREADS) {
        *(float4*)&out[rowBase + t0 + 4 * v] = *(const float4*)&cur[HALO + 4 * v];
    }
}

extern "C" void kernel_launch(void* const* d_in, const int* in_sizes, int n_in,
                              void* d_out, int out_size, void* d_ws, size_t ws_size,
                              hipStream_t stream) {
    (void)n_in; (void)d_ws; (void)ws_size; (void)out_size;
    const float* x  = (const float*)d_in[0];
    float* out      = (float*)d_out;

    const int L    = 262144;                 // row length (from reference shape)
    const int ROWS = in_sizes[0] / L;        // 16 * 8 = 128
    dim3 grid(L / T_TILE, ROWS);             // (128, 128) workgroups
    morph_co_51_25_kernel<<<grid, NTHREADS, 0, stream>>>(x, out, L);
}